// GraphConvolution_22660247454030
// MI455X (gfx1250) — compile-verified
//
#include <hip/hip_runtime.h>
#include <stdint.h>

typedef float v2f __attribute__((ext_vector_type(2)));
typedef float v8f __attribute__((ext_vector_type(8)));

#define IN_F 256
#define OUT_F 64
#define TILE_ROWS 128                    // rows of h per block (8 waves x 16 rows)
#define KCHUNK 32                        // K staged per pipeline stage
#define NCHUNKS (IN_F / KCHUNK)          // 8
#define XS_STRIDE 36                     // dwords/row: %4==0 (16B align), conflict-free A reads
#define XS_SIZE (TILE_ROWS * XS_STRIDE)  // 4608 dwords = 18432 B per buffer
#define WP_SIZE ((KCHUNK / 4) * 64 * 4)  // 8 kk-groups * 64 n * (2 half * 2 pair) = 2048 dwords

// --- CDNA5 async global->LDS copy (ASYNCcnt-tracked, no VGPR round trip) ----
__device__ __forceinline__ unsigned lds_off_u32(const void* p) {
    // flat LDS aperture address: low 32 bits == LDS byte offset (ISA 10.2)
    return (unsigned)(uintptr_t)p;
}
__device__ __forceinline__ void async_load_b128(unsigned lds_byte_off, const void* gptr) {
    asm volatile("global_load_async_to_lds_b128 %0, %1, off"
                 :: "v"(lds_byte_off), "v"((unsigned long long)(uintptr_t)gptr)
                 : "memory");
}
__device__ __forceinline__ void wait_async0() {
    asm volatile("s_wait_asynccnt 0x0" ::: "memory");
}

// ---------------------------------------------------------------------------
// Kernel 1: zero d_out (harness poisons it with 0xAA before timing)
// ---------------------------------------------------------------------------
__global__ void zero_kernel(float4* __restrict__ out, int n_f4) {
    int i = blockIdx.x * blockDim.x + threadIdx.x;
    if (i < n_f4) out[i] = make_float4(0.f, 0.f, 0.f, 0.f);
}

// ---------------------------------------------------------------------------
// Kernel 2: h = x @ W + b  via V_WMMA_F32_16X16X4_F32, double-buffered async
//   block = 256 threads = 8 waves; block computes h[row0..row0+127][0..63]
// ---------------------------------------------------------------------------
__global__ __launch_bounds__(256) void gemm_wmma_f32(
    const float* __restrict__ x,    // [N, 256]
    const float* __restrict__ W,    // [256, 64]
    const float* __restrict__ bias, // [64]
    float* __restrict__ h,          // [N, 64]
    int nRows)
{
    __shared__ __align__(16) float xs[2][XS_SIZE]; // 2 x 18432 B
    __shared__ __align__(16) float wp[2][WP_SIZE]; // 2 x  8192 B  (swizzled B frags)

    const int t    = threadIdx.x;
    const int lane = t & 31;
    const int wave = t >> 5;
    const int row0 = blockIdx.x * TILE_ROWS;
    const int half = lane >> 4;   // 0: lanes 0-15, 1: lanes 16-31
    const int lm   = lane & 15;

    // ---- staging helpers (inlined) ----
    auto stage_x = [&](float* xbuf, int kc) {
        #pragma unroll
        for (int i = 0; i < 4; ++i) {
            int idx = t + i * 256;            // 0..1023 float4 slots
            int r   = idx >> 3;               // 0..127
            int cq  = (idx & 7) << 2;         // 0..28
            int gr  = row0 + r;
            if (gr >= nRows) gr = nRows - 1;  // clamp: keep EXEC uniform, rows unguarded
            async_load_b128(lds_off_u32(&xbuf[r * XS_STRIDE + cq]),
                            x + (size_t)gr * IN_F + kc + cq);
        }
    };
    // W chunk, pre-swizzled to WMMA B layout:
    //   dword offset = (kkg*64 + n)*4 + half*2 + pair,  kb = kkg*4 + 2*half
    auto stage_w = [&](float* wbuf, int kc) {
        #pragma unroll
        for (int i = 0; i < 4; ++i) {
            int p  = t + i * 256;             // 0..1023 pairs
            int pr = p >> 6;                  // 0..15   (K pair-row: kb = 2*pr)
            int n  = p & 63;
            float a0 = W[(size_t)(kc + 2 * pr) * OUT_F + n];
            float a1 = W[(size_t)(kc + 2 * pr + 1) * OUT_F + n];
            float2 v = make_float2(a0, a1);
            *(float2*)&wbuf[(((pr >> 1) * 64 + n) << 2) + ((pr & 1) << 1)] = v;
        }
    };

    v8f acc[4];
    #pragma unroll
    for (int j = 0; j < 4; ++j) acc[j] = (v8f){0.f,0.f,0.f,0.f,0.f,0.f,0.f,0.f};

    // ---- prologue: stage chunk 0 ----
    stage_x(xs[0], 0);
    stage_w(wp[0], 0);
    wait_async0();
    __syncthreads();

    // ---- pipelined main loop: stage c+1 while computing c ----
    for (int c = 0; c < NCHUNKS; ++c) {
        const int cur = c & 1;
        if (c + 1 < NCHUNKS) {
            stage_x(xs[1 - cur], (c + 1) * KCHUNK);
            stage_w(wp[1 - cur], (c + 1) * KCHUNK);
        }

        const float* xrow = &xs[cur][(wave * 16 + lm) * XS_STRIDE];
        const float* wbuf = wp[cur];
        #pragma unroll
        for (int kk = 0; kk < KCHUNK; kk += 4) {
            int kb = kk + 2 * half;
            v2f a;
            a.x = xrow[kb];
            a.y = xrow[kb + 1];
            const float* wbase = &wbuf[(kk >> 2) * 256 + (half << 1)];
            #pragma unroll
            for (int j = 0; j < 4; ++j) {
                int n = j * 16 + lm;
                float2 bt = *(const float2*)&wbase[n << 2];
                v2f bf;
                bf.x = bt.x;
                bf.y = bt.y;
                acc[j] = __builtin_amdgcn_wmma_f32_16x16x4_f32(
                    /*neg_a=*/false, a, /*neg_b=*/false, bf,
                    /*c_mod=*/(short)0, acc[j],
                    /*reuse_a=*/false, /*reuse_b=*/false);
            }
        }

        wait_async0();      // chunk c+1 x-tile landed in LDS
        __syncthreads();    // all waves done reading cur + staging next
    }

    // ---- epilogue: add bias, store C (VGPR i -> row i + 8*half) ----
    #pragma unroll
    for (int j = 0; j < 4; ++j) {
        int n = j * 16 + lm;
        float bv = bias[n];
        #pragma unroll
        for (int i = 0; i < 8; ++i) {
            int gr = row0 + wave * 16 + i + 8 * half;
            if (gr < nRows)
                h[(size_t)gr * OUT_F + n] = acc[j][i] + bv;
        }
    }
}

// ---------------------------------------------------------------------------
// Kernel 3: out[row[e]] += val[e] * h[col[e]]   (16 threads per edge, f32x4)
//   h (25.6 MB) and out (25.6 MB) are L2-resident (192 MB L2): the random
//   gather + fp32 atomic scatter resolve in L2, not HBM.
// ---------------------------------------------------------------------------
__global__ __launch_bounds__(256) void scatter_kernel(
    const float* __restrict__ h,
    const int* __restrict__ edge_row,
    const int* __restrict__ edge_col,
    const float* __restrict__ edge_val,
    float* __restrict__ out,
    int nEdges)
{
    long long gid = (long long)blockIdx.x * blockDim.x + threadIdx.x;
    int e = (int)(gid >> 4);
    if (e >= nEdges) return;
    int f4 = ((int)gid & 15) << 2;     // feature base 0..60

    int   r = edge_row[e];
    int   c = edge_col[e];
    float v = edge_val[e];

    float4 hv = *(const float4*)(h + (size_t)c * OUT_F + f4);
    float* o  = out + (size_t)r * OUT_F + f4;
    unsafeAtomicAdd(o + 0, v * hv.x);
    unsafeAtomicAdd(o + 1, v * hv.y);
    unsafeAtomicAdd(o + 2, v * hv.z);
    unsafeAtomicAdd(o + 3, v * hv.w);
}

// ---------------------------------------------------------------------------
extern "C" void kernel_launch(void* const* d_in, const int* in_sizes, int n_in,
                              void* d_out, int out_size, void* d_ws, size_t ws_size,
                              hipStream_t stream) {
    const float* x  = (const float*)d_in[0];
    const int*   er = (const int*)d_in[1];
    const int*   ec = (const int*)d_in[2];
    const float* ev = (const float*)d_in[3];
    const float* W  = (const float*)d_in[4];
    const float* b  = (const float*)d_in[5];
    float*       out = (float*)d_out;
    float*       h   = (float*)d_ws;            // [nNodes, 64] = 25.6 MB

    const int nNodes = in_sizes[0] / IN_F;
    const int nEdges = in_sizes[1];

    // 1) zero output (poisoned by harness)
    int n_f4 = (nNodes * OUT_F) / 4;
    zero_kernel<<<(n_f4 + 255) / 256, 256, 0, stream>>>((float4*)out, n_f4);

    // 2) h = x @ W + b  (WMMA f32, async double-buffered)
    int gemmBlocks = (nNodes + TILE_ROWS - 1) / TILE_ROWS;
    gemm_wmma_f32<<<gemmBlocks, 256, 0, stream>>>(x, W, b, h, nNodes);

    // 3) edge scatter-add
    long long scatterThreads = (long long)nEdges * 16;
    int scatterBlocks = (int)((scatterThreads + 255) / 256);
    scatter_kernel<<<scatterBlocks, 256, 0, stream>>>(h, er, ec, ev, out, nEdges);
}